// SupervisedDiffPool_51788715655370
// MI455X (gfx1250) — compile-verified
//
#include <hip/hip_runtime.h>
#include <math.h>

typedef float v2f __attribute__((ext_vector_type(2)));
typedef float v8f __attribute__((ext_vector_type(8)));

#define BN_EPS 1e-5f

// ---------------------------------------------------------------------------
// Edge-parallel scatter-accumulate: agg[dst] += x[src]; cnt[dst] += 1.
// Each thread handles one (edge, 4-float chunk). chunks = F/4 = 1<<lgChunks.
// Hardware fp32 atomics (L2-side global_atomic_add_f32) — edge lists are the
// bandwidth/atomic bound of this workload, everything is L2-resident.
// ---------------------------------------------------------------------------
__global__ void scatter_accum_kernel(const int* __restrict__ ei, int E,
                                     const float* __restrict__ x, int F,
                                     float* __restrict__ agg,
                                     float* __restrict__ cnt,
                                     int lgChunks) {
    long tid = (long)blockIdx.x * blockDim.x + threadIdx.x;
    long total = (long)E << lgChunks;
    if (tid >= total) return;
    int e = (int)(tid >> lgChunks);
    int c = (int)(tid & ((1 << lgChunks) - 1));
    int src = ei[e];
    int dst = ei[E + e];
    const float4 v = *(const float4*)(x + (long)src * F + c * 4);
    float* ap = agg + (long)dst * F + c * 4;
    unsafeAtomicAdd(ap + 0, v.x);
    unsafeAtomicAdd(ap + 1, v.y);
    unsafeAtomicAdd(ap + 2, v.z);
    unsafeAtomicAdd(ap + 3, v.w);
    if (c == 0) unsafeAtomicAdd(cnt + dst, 1.0f);
}

// mean_nb = agg / max(cnt, 1)   (in place)
__global__ void seg_mean_div_kernel(float* __restrict__ agg,
                                    const float* __restrict__ cnt,
                                    long total, int lgF) {
    long tid = (long)blockIdx.x * blockDim.x + threadIdx.x;
    if (tid >= total) return;
    int row = (int)(tid >> lgF);
    agg[tid] = agg[tid] / fmaxf(cnt[row], 1.0f);
}

// ---------------------------------------------------------------------------
// Dual-input GEMM + bias + ReLU using fp32 WMMA:
//   C[M,N] = relu(A1[M,K1] @ W1[K1,N] + A2[M,K2] @ W2[K2,N] + bias[N])
// One wave per 16x16 output tile, K stepped by 4 via V_WMMA_F32_16X16X4_F32.
// All branching is wave-uniform => EXEC is all-ones at every WMMA.
//
// A frag (16x4 f32): lanes 0-15 carry K=0,1 ; lanes 16-31 carry K=2,3.
// B frag (4x16 f32): lane n holds column n; same K split across half-waves.
// C/D frag (16x16 f32): 8 VGPRs; lanes 0-15 -> M=r, lanes 16-31 -> M=8+r.
// ---------------------------------------------------------------------------
__global__ void wmma_dual_gemm_bias_relu(const float* __restrict__ A1, int K1,
                                         const float* __restrict__ A2, int K2,
                                         const float* __restrict__ W1,
                                         const float* __restrict__ W2,
                                         const float* __restrict__ bias,
                                         float* __restrict__ C, int M, int N) {
    const int wavesPerBlock = blockDim.x >> 5;
    const int wave = blockIdx.x * wavesPerBlock + (threadIdx.x >> 5);
    const int lane = threadIdx.x & 31;
    const int ntiles = N >> 4;
    const int mtiles = M >> 4;
    if (wave >= mtiles * ntiles) return;           // wave-uniform
    const int m0 = (wave / ntiles) << 4;
    const int n0 = (wave % ntiles) << 4;

    const int hl = lane & 15;        // row (A) / col (B) within tile
    const int hi = lane >> 4;        // half-wave selects K pair 0/1 vs 2/3
    const int arow = m0 + hl;
    const int bcol = n0 + hl;

    v8f acc = {};
    const float* a1p = A1 + (long)arow * K1 + 2 * hi;
    for (int k = 0; k < K1; k += 4) {
        v2f a, b;
        a.x = a1p[k + 0];
        a.y = a1p[k + 1];
        b.x = W1[(long)(k + 2 * hi + 0) * N + bcol];
        b.y = W1[(long)(k + 2 * hi + 1) * N + bcol];
        acc = __builtin_amdgcn_wmma_f32_16x16x4_f32(false, a, false, b,
                                                    (short)0, acc, false, false);
    }
    const float* a2p = A2 + (long)arow * K2 + 2 * hi;
    for (int k = 0; k < K2; k += 4) {
        v2f a, b;
        a.x = a2p[k + 0];
        a.y = a2p[k + 1];
        b.x = W2[(long)(k + 2 * hi + 0) * N + bcol];
        b.y = W2[(long)(k + 2 * hi + 1) * N + bcol];
        acc = __builtin_amdgcn_wmma_f32_16x16x4_f32(false, a, false, b,
                                                    (short)0, acc, false, false);
    }

    // Epilogue: + bias, ReLU, store. C/D layout: lane half selects M base.
    const int ccol = n0 + hl;
    const float bc = bias[ccol];
    const int rbase = m0 + hi * 8;
#pragma unroll
    for (int r = 0; r < 8; ++r) {
        float v = acc[r] + bc;
        v = fmaxf(v, 0.0f);
        C[(long)(rbase + r) * N + ccol] = v;
    }
}

// Per-column sum / sum-of-squares over M rows (N == blockDim.x == 256).
__global__ void col_stats_kernel(const float* __restrict__ h, int M, int N,
                                 float* __restrict__ cs, float* __restrict__ css) {
    int col = threadIdx.x;
    float s = 0.0f, s2 = 0.0f;
    for (int r = blockIdx.x; r < M; r += gridDim.x) {
        float v = h[(long)r * N + col];
        s += v;
        s2 += v * v;
    }
    unsafeAtomicAdd(&cs[col], s);
    unsafeAtomicAdd(&css[col], s2);
}

// BatchNorm (training stats, biased var) applied elementwise, N = 256.
__global__ void bn_apply_kernel(float* __restrict__ h, int M,
                                const float* __restrict__ cs,
                                const float* __restrict__ css,
                                const float* __restrict__ g,
                                const float* __restrict__ be) {
    long tid = (long)blockIdx.x * blockDim.x + threadIdx.x;
    if (tid >= (long)M * 256) return;
    int col = (int)(tid & 255);
    float invM = 1.0f / (float)M;
    float mu = cs[col] * invM;
    float var = css[col] * invM - mu * mu;
    h[tid] = (h[tid] - mu) * rsqrtf(var + BN_EPS) * g[col] + be[col];
}

// Fused BatchNorm + row-wise log_softmax; one block (256 thr) per row.
__global__ void bn_logsoftmax_kernel(float* __restrict__ h, int M,
                                     const float* __restrict__ cs,
                                     const float* __restrict__ css,
                                     const float* __restrict__ g,
                                     const float* __restrict__ be) {
    __shared__ float red[256];
    const int row = blockIdx.x;
    const int t = threadIdx.x;
    float invM = 1.0f / (float)M;
    float mu = cs[t] * invM;
    float var = css[t] * invM - mu * mu;
    float v = (h[(long)row * 256 + t] - mu) * rsqrtf(var + BN_EPS) * g[t] + be[t];

    red[t] = v;
    __syncthreads();
    for (int s = 128; s > 0; s >>= 1) {
        if (t < s) red[t] = fmaxf(red[t], red[t + s]);
        __syncthreads();
    }
    float mx = red[0];
    __syncthreads();
    red[t] = __expf(v - mx);
    __syncthreads();
    for (int s = 128; s > 0; s >>= 1) {
        if (t < s) red[t] += red[t + s];
        __syncthreads();
    }
    float lse = mx + logf(red[0]);
    h[(long)row * 256 + t] = v - lse;
}

// ---------------------------------------------------------------------------
// Host launcher. Only the emb branch reaches the output (pool branch is dead
// code in the reference), so we compute: scatter-mean -> SAGE GEMM -> BN,
// twice, then log_softmax into d_out.
// ---------------------------------------------------------------------------
extern "C" void kernel_launch(void* const* d_in, const int* in_sizes, int n_in,
                              void* d_out, int out_size, void* d_ws, size_t ws_size,
                              hipStream_t stream) {
    (void)in_sizes; (void)n_in; (void)out_size; (void)ws_size;

    const int N0 = 120000, N1 = 40000, N2 = 10000;
    const int E0 = 1280000, E1 = 320000;
    const int C_IN = 64, H = 256;
    (void)N0;

    const float* x   = (const float*)d_in[0];
    const int*   ei0 = (const int*)d_in[1];
    const int*   ei1 = (const int*)d_in[2];
    // emb branch weights: indices 13..22 (pool branch 3..12 is dead code)
    const float* Wl0 = (const float*)d_in[13];
    const float* Wr0 = (const float*)d_in[14];
    const float* b0  = (const float*)d_in[15];
    const float* g0  = (const float*)d_in[16];
    const float* be0 = (const float*)d_in[17];
    const float* Wl1 = (const float*)d_in[18];
    const float* Wr1 = (const float*)d_in[19];
    const float* b1  = (const float*)d_in[20];
    const float* g1  = (const float*)d_in[21];
    const float* be1 = (const float*)d_in[22];

    float* out = (float*)d_out;

    // Workspace layout (floats). Accumulator region first -> one memset.
    float* ws = (float*)d_ws;
    const long OFF_AGG0 = 0;                              // 40000*64
    const long OFF_CNT0 = OFF_AGG0 + (long)N1 * C_IN;     // 40000
    const long OFF_AGG1 = OFF_CNT0 + N1;                  // 10000*256
    const long OFF_CNT1 = OFF_AGG1 + (long)N2 * H;        // 10000
    const long OFF_STAT = OFF_CNT1 + N2;                  // 4*256
    const long OFF_H1   = OFF_STAT + 4 * 256;             // 40000*256
    float* agg0 = ws + OFF_AGG0;
    float* cnt0 = ws + OFF_CNT0;
    float* agg1 = ws + OFF_AGG1;
    float* cnt1 = ws + OFF_CNT1;
    float* cs0  = ws + OFF_STAT;
    float* css0 = cs0 + 256;
    float* cs1  = css0 + 256;
    float* css1 = cs1 + 256;
    float* h1   = ws + OFF_H1;

    // Zero all atomic accumulators (graph-capture-safe memset node).
    hipMemsetAsync(d_ws, 0, (size_t)OFF_H1 * sizeof(float), stream);

    // ---- Layer 0: segment-mean over ei0, SAGE GEMM, BN ----
    {
        long total = (long)E0 << 4;                        // 64 feats / 4
        int blocks = (int)((total + 255) / 256);
        scatter_accum_kernel<<<blocks, 256, 0, stream>>>(ei0, E0, x, C_IN,
                                                         agg0, cnt0, 4);
    }
    {
        long total = (long)N1 * C_IN;
        int blocks = (int)((total + 255) / 256);
        seg_mean_div_kernel<<<blocks, 256, 0, stream>>>(agg0, cnt0, total, 6);
    }
    {
        int waves = (N1 >> 4) * (H >> 4);                  // 2500*16
        int blocks = (waves + 7) / 8;
        wmma_dual_gemm_bias_relu<<<blocks, 256, 0, stream>>>(
            agg0, C_IN, x, C_IN, Wl0, Wr0, b0, h1, N1, H);
    }
    col_stats_kernel<<<512, 256, 0, stream>>>(h1, N1, H, cs0, css0);
    {
        long total = (long)N1 * H;
        int blocks = (int)((total + 255) / 256);
        bn_apply_kernel<<<blocks, 256, 0, stream>>>(h1, N1, cs0, css0, g0, be0);
    }

    // ---- Layer 1: segment-mean over ei1, SAGE GEMM, BN + log_softmax ----
    {
        long total = (long)E1 << 6;                        // 256 feats / 4
        int blocks = (int)((total + 255) / 256);
        scatter_accum_kernel<<<blocks, 256, 0, stream>>>(ei1, E1, h1, H,
                                                         agg1, cnt1, 6);
    }
    {
        long total = (long)N2 * H;
        int blocks = (int)((total + 255) / 256);
        seg_mean_div_kernel<<<blocks, 256, 0, stream>>>(agg1, cnt1, total, 8);
    }
    {
        int waves = (N2 >> 4) * (H >> 4);                  // 625*16
        int blocks = (waves + 7) / 8;
        wmma_dual_gemm_bias_relu<<<blocks, 256, 0, stream>>>(
            agg1, H, h1, H, Wl1, Wr1, b1, out, N2, H);
    }
    col_stats_kernel<<<512, 256, 0, stream>>>(out, N2, H, cs1, css1);
    bn_logsoftmax_kernel<<<N2, 256, 0, stream>>>(out, N2, cs1, css1, g1, be1);
}